// DiffGraphTransformerGenGCN_17703855194476
// MI455X (gfx1250) — compile-verified
//
#include <hip/hip_runtime.h>
#include <math.h>

// ---------------- problem constants ----------------
#define Bb     64
#define Nn     256
#define Hh     8
#define DHd    64
#define Dd     512        // H*DH
#define TOTAL  16384      // B*N
#define NT     131072     // H*TOTAL
#define Ee     262144     // TOTAL*DEG
#define ORD    4

#define FLAG_INIT 1
#define FLAG_BIAS 2

typedef float v2f __attribute__((ext_vector_type(2)));
typedef float v8f __attribute__((ext_vector_type(8)));

__device__ __forceinline__ v8f wmma_f32(v2f a, v2f b, v8f c) {
  // D = A(16x4) * B(4x16) + C, fp32, wave32
  return __builtin_amdgcn_wmma_f32_16x16x4_f32(false, a, false, b, (short)0, c,
                                               false, false);
}

// ---------------------------------------------------------------------------
// K1: per-graph (g = h*B + b) GCN stats -> Chebyshev coefficients coeff[512][4]
// ---------------------------------------------------------------------------
__global__ __launch_bounds__(256) void k_coeff(
    const float* __restrict__ attn, const float* __restrict__ Wg,
    const float* __restrict__ bg, const float* __restrict__ Wl,
    const float* __restrict__ bl, float* __restrict__ coeff) {
  const int g = blockIdx.x;       // g = h*B + b
  const int h = g >> 6;           // / B
  const int b = g & 63;
  const int j = threadIdx.x;      // column 0..255
  const float* Ag = attn + ((size_t)(b * Hh + h) * Nn) * Nn;  // A[i][j]

  __shared__ float sdinv[Nn];
  __shared__ float sred[Nn];

  // pass 1: column degree of A_hat = A + I
  float sum = 0.f;
  for (int i = 0; i < Nn; ++i) sum += Ag[i * Nn + j];
  float deg  = sum + 1.0f;
  float dinv = (deg > 0.f) ? rsqrtf(deg) : 0.f;
  sdinv[j] = dinv;
  __syncthreads();

  // pass 2: s[j] = (sum_i dinv[i]*A[i][j] + dinv[j]) * dinv[j]
  float t2 = 0.f;
  for (int i = 0; i < Nn; ++i) t2 += sdinv[i] * Ag[i * Nn + j];
  const float s = (t2 + dinv) * dinv;

  float gap[ORD];
  for (int c = 0; c < ORD; ++c) {
    const float wr = Wg[c] + Wg[4 + c] + Wg[8 + c] + Wg[12 + c]; // W_gcn.sum(0)
    float v = tanhf(s * wr + bg[c]);
    sred[j] = v;
    __syncthreads();
    for (int off = 128; off > 0; off >>= 1) {
      if (j < off) sred[j] += sred[j + off];
      __syncthreads();
    }
    if (j == 0) gap[c] = sred[0] * (1.0f / 256.0f);
    __syncthreads();
  }
  if (j == 0) {
    for (int c2 = 0; c2 < ORD; ++c2) {
      float acc = bl[c2];
      for (int c = 0; c < ORD; ++c) acc += gap[c] * Wl[c * ORD + c2];
      coeff[g * ORD + c2] = acc;
    }
  }
}

// ---------------------------------------------------------------------------
// generic: y[i] = alpha * x[i]  (alpha==0 -> pure zero fill, x not trusted)
// ---------------------------------------------------------------------------
__global__ void k_scale_set(float* __restrict__ y, const float* __restrict__ x,
                            float alpha, int n) {
  int i = blockIdx.x * blockDim.x + threadIdx.x;
  if (i < n) y[i] = (alpha == 0.f) ? 0.f : alpha * x[i];
}

// ---------------------------------------------------------------------------
// in-degree of base graph (16384 nodes, 262144 edges), then dinv_n = rsqrt
// ---------------------------------------------------------------------------
__global__ void k_degcount(const int* __restrict__ ei, float* __restrict__ deg) {
  int e = blockIdx.x * blockDim.x + threadIdx.x;
  if (e < Ee) atomicAdd(&deg[ei[Ee + e]], 1.0f);
}

__global__ void k_dinv(const float* __restrict__ deg, float* __restrict__ dinvn) {
  int v = blockIdx.x * blockDim.x + threadIdx.x;
  if (v < TOTAL) {
    float d = deg[v];
    dinvn[v] = (d > 0.f) ? rsqrtf(d) : 0.f;
  }
}

// ---------------------------------------------------------------------------
// x0[h*TOTAL + t][dh] = out_each_head[b][n][h][dh]   (t = b*N + n)
// ---------------------------------------------------------------------------
__global__ void k_x0(const float* __restrict__ oeh, float* __restrict__ x0) {
  int idx = blockIdx.x * blockDim.x + threadIdx.x;  // < NT*64
  int dh = idx & 63;
  int v  = idx >> 6;
  int h  = v >> 14;        // / TOTAL
  int t  = v & 16383;
  x0[idx] = oeh[(t * Hh + h) * DHd + dh];
}

// ---------------------------------------------------------------------------
// scatter prop: y[h*TOTAL+dst][dh] += scale * norm_e * x[h*TOTAL+src][dh]
// norm_e = -(dinv[src]*dinv[dst]) shared across heads
// ---------------------------------------------------------------------------
__global__ void k_prop(const int* __restrict__ ei, const float* __restrict__ dinvn,
                       const float* __restrict__ x, float* __restrict__ y,
                       float scale) {
  int idx = blockIdx.x * blockDim.x + threadIdx.x;  // < E*H*64 = 134217728
  int dh   = idx & 63;
  int rest = idx >> 6;
  int h    = rest & 7;
  int e    = rest >> 3;
  int s = ei[e];
  int d = ei[Ee + e];
  float nrm = -(dinvn[s] * dinvn[d]) * scale;
  float val = nrm * x[((h << 14) + s) * DHd + dh];
  atomicAdd(&y[((h << 14) + d) * DHd + dh], val);
}

// ---------------------------------------------------------------------------
// WMMA #1: filt[r][h*64+dh] (+)= ( c_node,k ⊙ T ) @ W_cheb[k]   [+ b_cheb]
// M = NT rows (v = h*TOTAL + t), K = 64, N = 64. Block = 8 waves x 16 rows.
// Output row permuted: r = n*B + b  (t = b*N + n).
// FLAGS is a compile-time template -> straight-line epilogue, no branches.
// ---------------------------------------------------------------------------
template <int FLAGS>
__global__ __launch_bounds__(256) void k_cheb(
    const float* __restrict__ T, const float* __restrict__ Wcheb,
    const float* __restrict__ coeff, const float* __restrict__ bcheb,
    float* __restrict__ filt, int kcheb) {
  const int wave = threadIdx.x >> 5;
  const int lane = threadIdx.x & 31;
  const int hlf  = lane >> 4;
  const int r    = lane & 15;
  const int vBase = blockIdx.x * 128 + wave * 16;
  const float* Wc = Wcheb + kcheb * DHd * DHd;

  // per-lane Chebyshev coefficient for its A-fragment row
  const int row0 = vBase + r;
  const int h0 = row0 >> 14;
  const int b0 = (row0 & 16383) >> 8;
  const float cA = coeff[(h0 * Bb + b0) * ORD + kcheb];
  const float* Trow = T + (size_t)row0 * DHd;

  v8f acc[4] = {};
  for (int kb = 0; kb < DHd; kb += 4) {
    v2f a;
    a.x = cA * Trow[kb + 2 * hlf];
    a.y = cA * Trow[kb + 2 * hlf + 1];
    const float* W0 = Wc + (kb + 2 * hlf) * DHd;
    const float* W1 = W0 + DHd;
#pragma unroll
    for (int nt = 0; nt < 4; ++nt) {
      v2f bv;
      bv.x = W0[nt * 16 + r];
      bv.y = W1[nt * 16 + r];
      acc[nt] = wmma_f32(a, bv, acc[nt]);
    }
  }

  // store with row permutation; tiles are block-exclusive -> plain RMW
#pragma unroll
  for (int j = 0; j < 8; ++j) {
    const int m   = j + 8 * hlf;
    const int row = vBase + m;
    const int h   = row >> 14;
    const int t   = row & 16383;
    const int bb  = t >> 8;
    const int n   = t & 255;
    const size_t rr = (size_t)n * Bb + bb;  // = n*B + b
#pragma unroll
    for (int nt = 0; nt < 4; ++nt) {
      const int dh  = nt * 16 + r;
      float val = acc[nt][j];
      if (FLAGS & FLAG_BIAS) val += bcheb[dh];
      const size_t o = rr * Dd + h * DHd + dh;
      if (FLAGS & FLAG_INIT)
        filt[o] = val;
      else
        filt[o] = filt[o] + val;
    }
  }
}

// ---------------------------------------------------------------------------
// WMMA #2 + fused LayerNorm:
// out[r][:] = LN( [output_row(r) | filt_row(r)] @ W_cat + b_cat )
// Block: 16 full rows x 512 cols (8 waves, each 16x64). K = 1024, KC = 64.
// A-panel staged via CDNA5 GLOBAL_LOAD_ASYNC_TO_LDS_B128 (ASYNCcnt-tracked),
// one b128 per thread per chunk; LDS row stride 68 floats = 272 B (16B-aligned
// rows, frag reads hit banks (r*4 [+2]) mod 64 -> conflict-free for 32 lanes).
// ---------------------------------------------------------------------------
__global__ __launch_bounds__(256) void k_gemm_ln(
    const float* __restrict__ Xout, const float* __restrict__ filt,
    const float* __restrict__ Wcat, const float* __restrict__ bcat,
    const float* __restrict__ gamma, const float* __restrict__ beta,
    float* __restrict__ out) {
  __shared__ __align__(16) float sA[16][68];  // 16B-aligned rows for b128
  __shared__ float sTile[16][Dd];             // 32 KB result tile for fused LN
  __shared__ float sSum[256], sSq[256];
  __shared__ float sMean[16], sRstd[16];

  const int tid  = threadIdx.x;
  const int wave = tid >> 5;
  const int lane = tid & 31;
  const int hlf  = lane >> 4;
  const int r    = lane & 15;
  const int rBase = blockIdx.x * 16;
  const int nBase = wave * 64;

  // per-thread staging slot: one float4 of the 16x64 chunk
  const int rw4 = tid >> 4;         // row 0..15
  const int kk  = (tid & 15) * 4;   // k offset 0..60
  const unsigned int ldsa =
      (unsigned int)(size_t)&sA[rw4][kk];  // flat->LDS: low 32 bits

  v8f acc[4] = {};
  for (int kc = 0; kc < 1024; kc += 64) {
    __syncthreads();  // previous chunk fully consumed before overwrite
    // chunk-uniform source select (512 boundary is a multiple of 64)
    const float* srcBase = (kc < Dd) ? (Xout + kc) : (filt + (kc - Dd));
    const float* gsrc = srcBase + (size_t)(rBase + rw4) * Dd + kk;
    asm volatile("global_load_async_to_lds_b128 %0, %1, off"
                 :: "v"(ldsa), "v"(gsrc) : "memory");
    asm volatile("s_wait_asynccnt 0x0" ::: "memory");
    __syncthreads();

    for (int ks = 0; ks < 64; ks += 4) {
      const int kG = kc + ks;
      v2f a;
      a.x = sA[r][ks + 2 * hlf];
      a.y = sA[r][ks + 2 * hlf + 1];
      const float* W0 = Wcat + (size_t)(kG + 2 * hlf) * Dd + nBase;
      const float* W1 = W0 + Dd;
#pragma unroll
      for (int nt = 0; nt < 4; ++nt) {
        v2f bv;
        bv.x = W0[nt * 16 + r];
        bv.y = W1[nt * 16 + r];
        acc[nt] = wmma_f32(a, bv, acc[nt]);
      }
    }
  }

  __syncthreads();
#pragma unroll
  for (int j = 0; j < 8; ++j) {
    const int m = j + 8 * hlf;
#pragma unroll
    for (int nt = 0; nt < 4; ++nt) {
      const int col = nBase + nt * 16 + r;
      sTile[m][col] = acc[nt][j] + bcat[col];
    }
  }
  __syncthreads();

  // per-row mean / variance over 512 columns (16 threads per row)
  {
    const int rowid = tid >> 4;
    const int sub   = tid & 15;
    float s = 0.f, ss = 0.f;
    for (int c = sub; c < Dd; c += 16) {
      const float v = sTile[rowid][c];
      s += v;
      ss += v * v;
    }
    sSum[tid] = s;
    sSq[tid]  = ss;
  }
  __syncthreads();
  if (tid < 16) {
    float s = 0.f, ss = 0.f;
    for (int i = 0; i < 16; ++i) {
      s  += sSum[tid * 16 + i];
      ss += sSq[tid * 16 + i];
    }
    const float mu  = s * (1.0f / 512.0f);
    const float var = ss * (1.0f / 512.0f) - mu * mu;
    sMean[tid] = mu;
    sRstd[tid] = rsqrtf(var + 1e-5f);
  }
  __syncthreads();

  for (int i = tid; i < 16 * Dd; i += 256) {
    const int rw = i >> 9, c = i & 511;
    const float v = (sTile[rw][c] - sMean[rw]) * sRstd[rw] * gamma[c] + beta[c];
    out[(size_t)(rBase + rw) * Dd + c] = v;
  }
}

// ---------------------------------------------------------------------------
extern "C" void kernel_launch(void* const* d_in, const int* in_sizes, int n_in,
                              void* d_out, int out_size, void* d_ws, size_t ws_size,
                              hipStream_t stream) {
  (void)in_sizes; (void)n_in; (void)out_size; (void)ws_size;
  const float* outp  = (const float*)d_in[0];   // output   (N,B,D)
  const float* attn  = (const float*)d_in[1];   // attn     (B,H,N,N)
  const float* oeh   = (const float*)d_in[2];   // out_each_head (B,N,H,DH)
  const int*   ei    = (const int*)d_in[3];     // edge_index (2,E) int32
  // d_in[4] feature_indices, d_in[5] batch: mappings derived analytically
  const float* Wg    = (const float*)d_in[6];
  const float* bg    = (const float*)d_in[7];
  const float* Wl    = (const float*)d_in[8];
  const float* bl    = (const float*)d_in[9];
  const float* Wch   = (const float*)d_in[10];
  const float* bch   = (const float*)d_in[11];
  const float* Wcat  = (const float*)d_in[12];
  const float* bcat  = (const float*)d_in[13];
  const float* gamma = (const float*)d_in[14];
  const float* beta  = (const float*)d_in[15];
  float* out = (float*)d_out;

  float* ws    = (float*)d_ws;
  float* Ta    = ws;                       // NT*64 = 8388608 floats
  float* Tb    = ws + 8388608;
  float* Tc    = ws + 16777216;
  float* filt  = ws + 25165824;            // 16384 x 512
  float* deg   = ws + 33554432;            // 16384
  float* dinvn = ws + 33570816;            // 16384
  float* coeff = ws + 33587200;            // 512 x 4

  const int TPB = 256;
  const int nT   = NT * DHd;               // 8388608
  const int gT   = nT / TPB;               // 32768
  const int gP   = (Ee * Hh * DHd) / TPB;  // 524288
  const int gCh  = NT / 128;               // 1024
  const int gLN  = TOTAL / 16;             // 1024

  // 1) graph stats -> Chebyshev coefficients
  k_coeff<<<Hh * Bb, TPB, 0, stream>>>(attn, Wg, bg, Wl, bl, coeff);

  // 2) node degrees -> dinv_n (base graph; identical across heads)
  k_scale_set<<<TOTAL / TPB, TPB, 0, stream>>>(deg, deg, 0.f, TOTAL);
  k_degcount<<<Ee / TPB, TPB, 0, stream>>>(ei, deg);
  k_dinv<<<TOTAL / TPB, TPB, 0, stream>>>(deg, dinvn);

  // 3) T0 = x0
  k_x0<<<gT, TPB, 0, stream>>>(oeh, Ta);
  k_cheb<FLAG_INIT><<<gCh, TPB, 0, stream>>>(Ta, Wch, coeff, bch, filt, 0);

  // 4) T1 = prop(T0)
  k_scale_set<<<gT, TPB, 0, stream>>>(Tb, Tb, 0.f, nT);
  k_prop<<<gP, TPB, 0, stream>>>(ei, dinvn, Ta, Tb, 1.0f);
  k_cheb<0><<<gCh, TPB, 0, stream>>>(Tb, Wch, coeff, bch, filt, 1);

  // 5) T2 = 2*prop(T1) - T0
  k_scale_set<<<gT, TPB, 0, stream>>>(Tc, Ta, -1.0f, nT);
  k_prop<<<gP, TPB, 0, stream>>>(ei, dinvn, Tb, Tc, 2.0f);
  k_cheb<0><<<gCh, TPB, 0, stream>>>(Tc, Wch, coeff, bch, filt, 2);

  // 6) T3 = 2*prop(T2) - T1   (reuse Ta)
  k_scale_set<<<gT, TPB, 0, stream>>>(Ta, Tb, -1.0f, nT);
  k_prop<<<gP, TPB, 0, stream>>>(ei, dinvn, Tc, Ta, 2.0f);
  k_cheb<FLAG_BIAS><<<gCh, TPB, 0, stream>>>(Ta, Wch, coeff, bch, filt, 3);

  // 7) big GEMM [output | filt] @ W_cat + bias, fused LayerNorm
  k_gemm_ln<<<gLN, TPB, 0, stream>>>(outp, filt, Wcat, bcat, gamma, beta, out);
}